// SkipGram_71116068487998
// MI455X (gfx1250) — compile-verified
//
#include <hip/hip_runtime.h>
#include <hip/hip_bf16.h>
#include <math.h>

typedef __attribute__((ext_vector_type(2))) float v2f;
typedef __attribute__((ext_vector_type(4))) float v4f;
typedef __attribute__((ext_vector_type(8))) float v8f;

#define VOCAB 100000
#define DIM 128
#define BATCH 4096
#define NEG 5
#define PL 6
#define TILE 16                         // batch rows per block
#define ROWS_PER_BLOCK (TILE * (2 + NEG))   // 112 pooled rows
#define LDS_STRIDE 132                  // 128 + 4 pad: stride%64==4 -> conflict-free WMMA frag loads

__device__ __forceinline__ float log_sigmoid(float x) {
    // numerically stable: min(x,0) - log1p(exp(-|x|))
    return fminf(x, 0.0f) - log1pf(expf(-fabsf(x)));
}

__global__ __launch_bounds__(256) void sg_fused(
    const float* __restrict__ u_emb, const float* __restrict__ v_emb,
    const int* __restrict__ pos_u_idx, const int* __restrict__ pos_u_len,
    const int* __restrict__ pos_v_idx, const int* __restrict__ pos_v_len,
    const int* __restrict__ neg_v_idx, const int* __restrict__ neg_v_len,
    float* __restrict__ ws)
{
    __shared__ float lds[ROWS_PER_BLOCK * LDS_STRIDE];   // 59,136 bytes
    const int t    = blockIdx.x;            // 16-batch tile index
    const int lane = threadIdx.x & 31;
    const int wave = threadIdx.x >> 5;      // 0..7 (8 waves of wave32)

    // ---------------- Phase 1: ragged mean-pool (memory-bound gather) -------------
    // One wave per pooled row; lane j handles float4 chunk j (32*4 = 128 = DIM).
    // 112 rows / 8 waves = 14 rows per wave. Row meaning:
    //   r in [0,16):   pooled_u   for batch t*16+r
    //   r in [16,32):  pooled_v   for batch t*16+(r-16)
    //   r in [32,112): pooled_neg for (t*16+i)*5+q, r = 32 + i*5 + q
    for (int r = wave * 14; r < wave * 14 + 14; ++r) {
        const float* table;
        const int*   ip;
        int n;
        if (r < TILE) {
            int b = t * TILE + r;
            table = u_emb; ip = pos_u_idx + b * PL; n = pos_u_len[b] + 1;
        } else if (r < 2 * TILE) {
            int b = t * TILE + (r - TILE);
            table = v_emb; ip = pos_v_idx + b * PL; n = pos_v_len[b] + 1;
        } else {
            int m = t * (TILE * NEG) + (r - 2 * TILE);
            table = v_emb; ip = neg_v_idx + m * PL; n = neg_v_len[m] + 1;
        }
        v4f acc = {};
        #pragma unroll
        for (int k = 0; k < PL; ++k) {
            if (k < n) {   // wave-uniform predicate
                const v4f* src = (const v4f*)(table + (size_t)ip[k] * DIM);
                acc += src[lane];          // 32 lanes x 16B = 512B coalesced row gather
            }
        }
        float inv = 1.0f / (float)n;
        acc *= inv;
        *(v4f*)(&lds[r * LDS_STRIDE + lane * 4]) = acc;
    }
    __syncthreads();

    // ---------------- Phase 2: WMMA f32 16x16x4 dot-product tiles -----------------
    // Wave 0: S = U x V^T (pos scores on diagonal)
    // Wave w=1..5: S = U x NegV_{w-1}^T (neg scores on diagonal)
    if (wave < 6) {   // wave-uniform branch: EXEC all-1s inside (WMMA requirement)
        const int m  = lane & 15;     // matrix row this lane addresses
        const int hi = lane >> 4;     // 0 -> K pair (k,k+1), 1 -> (k+2,k+3)
        const float* arow = &lds[m * LDS_STRIDE];                         // U row m
        const int brow_idx = (wave == 0) ? (TILE + m)
                                         : (2 * TILE + m * NEG + (wave - 1));
        const float* brow = &lds[brow_idx * LDS_STRIDE];                  // B[k][n] = vrow[n][k]

        v8f acc = {};
        #pragma unroll
        for (int kb = 0; kb < DIM; kb += 4) {
            const int koff = kb + (hi << 1);
            v2f a = *(const v2f*)(arow + koff);   // A frag: {u[m][koff], u[m][koff+1]}
            v2f b = *(const v2f*)(brow + koff);   // B frag: {v[m][koff], v[m][koff+1]}
            acc = __builtin_amdgcn_wmma_f32_16x16x4_f32(
                false, a, false, b, (short)0, acc, false, false);
        }

        // Diagonal extraction (C layout: VGPR r / lanes0-15: M=r; lanes16-31: M=r+8)
        //   lane l in [0,8):   S[l][l]       = acc[l]
        //   lane l in [24,32): S[l-16][l-16] = acc[l-24]
        const bool valid = (lane < 8) || (lane >= 24);
        const int sel = (lane < 8) ? lane : (lane - 24);
        float diag = acc[0];
        if (sel == 1) diag = acc[1];
        if (sel == 2) diag = acc[2];
        if (sel == 3) diag = acc[3];
        if (sel == 4) diag = acc[4];
        if (sel == 5) diag = acc[5];
        if (sel == 6) diag = acc[6];
        if (sel == 7) diag = acc[7];

        const float x  = (wave == 0) ? diag : -diag;
        float ls = valid ? log_sigmoid(x) : 0.0f;

        // wave32 butterfly reduction
        #pragma unroll
        for (int off = 16; off > 0; off >>= 1)
            ls += __shfl_xor(ls, off, 32);

        if (lane == 0) ws[t * 6 + wave] = ls;
    }
}

// Deterministic final reduction: 256 blocks x 6 partials = 1536 floats
__global__ __launch_bounds__(256) void sg_reduce(const float* __restrict__ ws,
                                                 float* __restrict__ out) {
    __shared__ float sm[256];
    float s = 0.0f;
    for (int i = threadIdx.x; i < (BATCH / TILE) * 6; i += 256) s += ws[i];
    sm[threadIdx.x] = s;
    __syncthreads();
    for (int stride = 128; stride > 0; stride >>= 1) {
        if (threadIdx.x < stride) sm[threadIdx.x] += sm[threadIdx.x + stride];
        __syncthreads();
    }
    if (threadIdx.x == 0) out[0] = -sm[0] / (float)BATCH;
}

extern "C" void kernel_launch(void* const* d_in, const int* in_sizes, int n_in,
                              void* d_out, int out_size, void* d_ws, size_t ws_size,
                              hipStream_t stream) {
    const float* u_emb  = (const float*)d_in[0];
    const float* v_emb  = (const float*)d_in[1];
    const int*   pu_idx = (const int*)d_in[2];
    const int*   pu_len = (const int*)d_in[3];
    const int*   pv_idx = (const int*)d_in[4];
    const int*   pv_len = (const int*)d_in[5];
    const int*   nv_idx = (const int*)d_in[6];
    const int*   nv_len = (const int*)d_in[7];
    float* ws = (float*)d_ws;

    sg_fused<<<BATCH / TILE, 256, 0, stream>>>(u_emb, v_emb, pu_idx, pu_len,
                                               pv_idx, pv_len, nv_idx, nv_len, ws);
    sg_reduce<<<1, 256, 0, stream>>>(ws, (float*)d_out);
}